// JointCandidate_40750649704563
// MI455X (gfx1250) — compile-verified
//
#include <hip/hip_runtime.h>

typedef __attribute__((ext_vector_type(2))) float v2f;
typedef __attribute__((ext_vector_type(8))) float v8f;

#define NB 8
#define NN 192
#define NC 16

// ---------------------------------------------------------------------------
// Kernel 1: argmax over last dim (first-max semantics, like jnp.argmax),
// store the winning index as float into workspace. 3*NB*NN = 4608 rows total.
// ---------------------------------------------------------------------------
__global__ void __launch_bounds__(256)
argmax_rows_kernel(const float* __restrict__ t0,
                   const float* __restrict__ t1,
                   const float* __restrict__ t2,
                   float* __restrict__ ws) {
  const int rows = NB * NN;                       // 1536 rows per tensor
  int tid = blockIdx.x * blockDim.x + threadIdx.x;
  if (tid >= 3 * rows) return;
  const float* src = (tid < rows) ? t0 : ((tid < 2 * rows) ? t1 : t2);
  int row = tid % rows;
  const float* p = src + row * NC;
  float best = p[0];
  int bi = 0;
#pragma unroll
  for (int c = 1; c < NC; ++c) {
    float v = p[c];
    if (v > best) { best = v; bi = c; }           // strict > => first max wins
  }
  ws[tid] = (float)bi;
}

// ---------------------------------------------------------------------------
// Kernel 2: streaming fill. One 4-wave block per (b, j, i-tile-group).
// Each wave owns a 16-row i-tile and sweeps 12 k-tiles of 16.
// Per tile: D(16x16) = outer(a0[i0..i0+15], a1[j]*a2[k0..k0+15]) via
// V_WMMA_F32_16X16X4_F32 (only K=0 populated), then distinct-mask + NT store.
// ---------------------------------------------------------------------------
__global__ void __launch_bounds__(128)
outer_mask_fill_kernel(const float* __restrict__ ws, float* __restrict__ out) {
  const int lane = threadIdx.x & 31;
  const int wave = threadIdx.x >> 5;

  int bid = blockIdx.x;                 // grid = NB * NN * 3
  int itg = bid % 3;
  int j   = (bid / 3) % NN;
  int b   = bid / (3 * NN);
  int i0  = (itg * 4 + wave) * 16;      // 12 i-tiles total across 3 groups x 4 waves

  const float* a0 = ws;
  const float* a1 = ws + NB * NN;
  const float* a2 = ws + 2 * NB * NN;

  const int lk = lane & 15;
  const int ihalf = (lane >> 4) << 3;   // lanes 16-31 hold rows M+8

  // A: 16x4 f32, only K=0 column nonzero (VGPR0 lanes 0-15), rest zero.
  float a0v = (lane < 16) ? a0[b * NN + i0 + lk] : 0.0f;
  v2f A; A.x = a0v; A.y = 0.0f;

  float a1j = a1[b * NN + j];

  // out[b][i][j][k] = out + b*N^3 + i*N^2 + j*N + k
  float* outbj = out + (size_t)b * NN * NN * NN + (size_t)j * NN;

  for (int k0 = 0; k0 < NN; k0 += 16) {
    // B: 4x16 f32, only K=0 row nonzero (VGPR0 lanes 0-15), rest zero.
    float a2v = (lane < 16) ? a2[b * NN + k0 + lk] : 0.0f;
    v2f Bm; Bm.x = a1j * a2v; Bm.y = 0.0f;

    v8f Cm = {};
    v8f D = __builtin_amdgcn_wmma_f32_16x16x4_f32(
        /*neg_a=*/false, A, /*neg_b=*/false, Bm,
        /*c_mod=*/(short)0, Cm, /*reuse_a=*/false, /*reuse_b=*/false);

    const int k = k0 + lk;
    const bool jk_ne = (j != k);
    float* p = outbj + (size_t)(i0 + ihalf) * (NN * NN) + k;

#pragma unroll
    for (int v = 0; v < 8; ++v) {
      const int i = i0 + ihalf + v;
      const bool keep = (i != j) && (i != k) && jk_ne;
      float val = keep ? D[v] : 0.0f;
      // 226 MB pure write stream > 192 MB L2: keep it non-temporal.
      __builtin_nontemporal_store(val, p + (size_t)v * (NN * NN));
    }
  }
}

// ---------------------------------------------------------------------------
extern "C" void kernel_launch(void* const* d_in, const int* in_sizes, int n_in,
                              void* d_out, int out_size, void* d_ws, size_t ws_size,
                              hipStream_t stream) {
  (void)in_sizes; (void)n_in; (void)out_size; (void)ws_size;
  const float* t0 = (const float*)d_in[0];
  const float* t1 = (const float*)d_in[1];
  const float* t2 = (const float*)d_in[2];
  float* ws  = (float*)d_ws;            // 3*1536 floats = 18 KB
  float* out = (float*)d_out;           // 8*192^3 floats

  const int rows3 = 3 * NB * NN;        // 4608
  argmax_rows_kernel<<<(rows3 + 255) / 256, 256, 0, stream>>>(t0, t1, t2, ws);

  const int grid = NB * NN * 3;         // 4608 blocks, 128 threads (4 waves) each
  outer_mask_fill_kernel<<<grid, 128, 0, stream>>>(ws, out);
}